// QuantGelu_90443421319772
// MI455X (gfx1250) — compile-verified
//
#include <hip/hip_runtime.h>

typedef float v4f __attribute__((ext_vector_type(4)));
typedef int   v4i __attribute__((ext_vector_type(4)));

#define DCH    4096              // channels per row (last axis)
#define NT     256               // threads per block = 8 wave32s
#define VEC    4                 // floats per 128-bit access
#define CHUNK  (NT * VEC)        // 1024 floats per block-wide chunk
#define NCHUNK (DCH / CHUNK)     // 4 chunks cover one row
#define NWAVE  (NT / 32)

typedef __attribute__((address_space(1))) v4i gv4i;   // global 16B vector
typedef __attribute__((address_space(3))) v4i lv4i;   // LDS 16B vector

// head of I-BERT ShiftExp: v * 1.4375 ~= v / ln2 via exact floor ops
__device__ __forceinline__ float ies_head(float v) {
    return v + floorf(v * 0.5f) - floorf(v * 0.0625f);
}

// tail: clamp, divide by x0 (reciprocal-multiply; t,x0 integer-valued and
// |t/x0|<=15, so eps=1e-5 makes the floor exact vs true division), then
// floor(e * 2^(15-q)) via v_ldexp_f32.
__device__ __forceinline__ float ies_tail(float t, float x0, float inv_x0) {
    t = fmaxf(t, 15.0f * x0);                       // x0 < 0: clamp from below
    float qf = floorf(fmaf(t, inv_x0, 1e-5f));
    float r  = fmaf(-x0, qf, t);                    // exact: small integers
    float e  = fmaf(r, 0.5f, -x0);
    return fmaxf(floorf(ldexpf(e, 15 - (int)qf)), 0.0f);
}

// one-time per-channel constants (row-invariant): 1/s, x0, 1/x0, s/128
__global__ __launch_bounds__(NT) void chan_precompute(
    const float* __restrict__ scaler,
    float* __restrict__ t_inv, float* __restrict__ t_x0,
    float* __restrict__ t_ix0, float* __restrict__ t_osc)
{
    const int d = blockIdx.x * NT + threadIdx.x;
    const float s  = scaler[d];
    const float x0 = floorf(-1.0f / (s * 1.702f));
    t_inv[d] = 1.0f / s;
    t_x0[d]  = x0;
    t_ix0[d] = 1.0f / x0;
    t_osc[d] = s * 0.0078125f;                      // scaler / 128
}

template <bool TAB>
__global__ __launch_bounds__(NT) void quantgelu_rows(
    const float* __restrict__ x,
    const float* __restrict__ scaler,
    const float* __restrict__ t_inv,
    const float* __restrict__ t_x0,
    const float* __restrict__ t_ix0,
    const float* __restrict__ t_osc,
    float* __restrict__ y)
{
    __shared__ __align__(16) float lds_x[DCH];   // 16 KB staged row
    __shared__ float lds_red[NWAVE];

    const int t = threadIdx.x;
    const size_t row = blockIdx.x;
    const float* xrow = x + row * (size_t)DCH;
    float*       yrow = y + row * (size_t)DCH;

    // ---- Stage the row global -> LDS via the async copy engine (ASYNCcnt).
    // x is streamed exactly once: cpol=1 -> th:TH_LOAD_NT keeps the 128 MB
    // stream from evicting the hot per-channel tables in WGP$/L2.
#if __has_builtin(__builtin_amdgcn_global_load_async_to_lds_b128)
    #pragma unroll
    for (int k = 0; k < NCHUNK; ++k) {
        __builtin_amdgcn_global_load_async_to_lds_b128(
            (gv4i*)(xrow + k * CHUNK + t * VEC),
            (lv4i*)(lds_x + k * CHUNK + t * VEC),
            /*offset=*/0, /*cpol=NT*/1);
    }
    asm volatile("s_wait_asynccnt 0" ::: "memory");
    // no barrier needed: each lane reads back only the bytes it staged,
    // and s_wait_asynccnt 0 covers this wave's async writes to LDS.
#else
    #pragma unroll
    for (int k = 0; k < NCHUNK; ++k)
        *(v4f*)(lds_x + k * CHUNK + t * VEC) =
            *(const v4f*)(xrow + k * CHUNK + t * VEC);
#endif

    // ---- Phase 1: per-channel quantize + per-thread max ----
    float pre[NCHUNK * VEC];
    float m = -3.4e38f;
    #pragma unroll
    for (int k = 0; k < NCHUNK; ++k) {
        const int base = k * CHUNK + t * VEC;
        v4f xv = *(v4f*)(lds_x + base);
        v4f iv;
        if (TAB) {
            iv = *(const v4f*)(t_inv + base);              // L2-hot table
        } else {
            v4f sv = *(const v4f*)(scaler + base);
            #pragma unroll
            for (int j = 0; j < VEC; ++j) iv[j] = 1.0f / sv[j];
        }
        #pragma unroll
        for (int j = 0; j < VEC; ++j) {
            float p = rintf(xv[j] * iv[j]);                // v_rndne: RNE like jnp.round
            p = fminf(fmaxf(p, -128.0f), 127.0f);          // v_med3 clip
            pre[k * VEC + j] = p;
            m = fmaxf(m, p);
        }
    }

    // ---- Row-max reduction: wave32 butterfly, then across the 8 waves ----
    #pragma unroll
    for (int off = 16; off > 0; off >>= 1)
        m = fmaxf(m, __shfl_xor(m, off, 32));
    if ((t & 31) == 0) lds_red[t >> 5] = m;
    __syncthreads();
    float rowmax = lds_red[0];
    #pragma unroll
    for (int i = 1; i < NWAVE; ++i) rowmax = fmaxf(rowmax, lds_red[i]);

    // exp_int_max head depends only on the row scalar: hoist out of the loop
    const float tm_head = ies_head(-rowmax);

    // ---- Phase 2: integer sigmoid, scale back, stream out ----
    #pragma unroll
    for (int k = 0; k < NCHUNK; ++k) {
        const int base = k * CHUNK + t * VEC;
        v4f x0v, ix0v, oscv;
        if (TAB) {
            x0v  = *(const v4f*)(t_x0  + base);
            ix0v = *(const v4f*)(t_ix0 + base);
            oscv = *(const v4f*)(t_osc + base);
        } else {
            v4f sv = *(const v4f*)(scaler + base);
            #pragma unroll
            for (int j = 0; j < VEC; ++j) {
                x0v[j]  = floorf(-1.0f / (sv[j] * 1.702f));
                ix0v[j] = 1.0f / x0v[j];
                oscv[j] = sv[j] * 0.0078125f;
            }
        }
        v4f out;
        #pragma unroll
        for (int j = 0; j < VEC; ++j) {
            const float x0 = x0v[j], ix0 = ix0v[j];
            const float p  = pre[k * VEC + j];
            const float ei = ies_tail(ies_head(p - rowmax), x0, ix0);
            const float em = ies_tail(tm_head, x0, ix0);
            const float sum    = fminf(ei + em, 2147483647.0f);
            const float factor = floorf(2147483647.0f / sum);   // keep precise
            const float sig    = floorf(ei * factor * (1.0f / 16777216.0f)); // /2^24
            out[j] = p * sig * oscv[j];
        }
        // output never re-read: bypass caches with non-temporal 128-bit stores
        __builtin_nontemporal_store(out, (v4f*)(yrow + base));
    }
}

extern "C" void kernel_launch(void* const* d_in, const int* in_sizes, int n_in,
                              void* d_out, int out_size, void* d_ws, size_t ws_size,
                              hipStream_t stream) {
    const float* x      = (const float*)d_in[0];   // [4, 2048, 4096] fp32
    const float* scaler = (const float*)d_in[1];   // [4096] fp32
    float* y = (float*)d_out;                      // [4, 2048, 4096] fp32

    const int rows = in_sizes[0] / DCH;            // 4*2048 = 8192 rows
    const size_t need = (size_t)4 * DCH * sizeof(float);   // 64 KB of tables

    if (ws_size >= need) {
        float* t_inv = (float*)d_ws;
        float* t_x0  = t_inv + DCH;
        float* t_ix0 = t_x0  + DCH;
        float* t_osc = t_ix0 + DCH;
        chan_precompute<<<dim3(DCH / NT), dim3(NT), 0, stream>>>(
            scaler, t_inv, t_x0, t_ix0, t_osc);
        quantgelu_rows<true><<<dim3(rows), dim3(NT), 0, stream>>>(
            x, scaler, t_inv, t_x0, t_ix0, t_osc, y);
    } else {
        quantgelu_rows<false><<<dim3(rows), dim3(NT), 0, stream>>>(
            x, scaler, nullptr, nullptr, nullptr, nullptr, y);
    }
}